// TextureBaker_77232101916817
// MI455X (gfx1250) — compile-verified
//
#include <hip/hip_runtime.h>
#include <math.h>

typedef __attribute__((ext_vector_type(2))) float v2f;
typedef __attribute__((ext_vector_type(8))) float v8f;

// ---------------------------------------------------------------------------
// Kernel 1: per-triangle affine barycentric coefficients.
//   w0(px,py) = a0*px + b0*py + c0,  w1(px,py) = a1*px + b1*py + c1
//   w2 = 1 - w0 - w1. Degenerate triangles (|denom| <= 1e-12) get c = -1 so
//   they can never pass the inside test (matches reference `valid` masking).
// ---------------------------------------------------------------------------
__global__ __launch_bounds__(256) void bake_prep_coeffs(
    const float* __restrict__ uv, const int* __restrict__ faces,
    float* __restrict__ coef, int nf, int nfPad)
{
    int t = blockIdx.x * blockDim.x + threadIdx.x;
    if (t >= nfPad) return;
    float a0 = 0.f, b0 = 0.f, c0 = -1.f;
    float a1 = 0.f, b1 = 0.f, c1 = -1.f;
    if (t < nf) {
        int i0 = faces[3 * t + 0], i1 = faces[3 * t + 1], i2 = faces[3 * t + 2];
        float ax = uv[2 * i0 + 0], ay = uv[2 * i0 + 1];
        float bx = uv[2 * i1 + 0], by = uv[2 * i1 + 1];
        float cx = uv[2 * i2 + 0], cy = uv[2 * i2 + 1];
        float denom = (by - cy) * (ax - cx) + (cx - bx) * (ay - cy);
        if (fabsf(denom) > 1e-12f) {
            float inv = 1.0f / denom;
            a0 = (by - cy) * inv;
            b0 = (cx - bx) * inv;
            c0 = -(a0 * cx + b0 * cy);
            a1 = (cy - ay) * inv;
            b1 = (ax - cx) * inv;
            c1 = -(a1 * cx + b1 * cy);
        }
    }
    float* o = coef + 6 * t;
    o[0] = a0; o[1] = b0; o[2] = c0;
    o[3] = a1; o[4] = b1; o[5] = c1;
}

// ---------------------------------------------------------------------------
// Kernel 2: WMMA rasterizer. Each wave32 owns 16 consecutive pixels.
// Per 16-triangle block:
//   A (16x4 f32, 2 VGPRs): lanes 0-15 hold (px,py) rows K=0,1;
//                          lanes 16-31 hold (1,0)  rows K=2,3.
//   B (4x16 f32, 2 VGPRs): lanes 0-15 hold (a,b) for tri=lane;
//                          lanes 16-31 hold (c,0) for tri=lane-16.
//   D = A*B: VGPR v, lanes 0-15 -> pixel v, lanes 16-31 -> pixel v+8,
//            column N = lane%16 = triangle-in-block.
// Per D-VGPR a single ballot gives the 16-tri hit masks for two pixels;
// ctz picks the lowest triangle index (= reference "first inside" priority).
// ---------------------------------------------------------------------------
__global__ __launch_bounds__(256) void bake_wmma_raster(
    const float* __restrict__ attr, const int* __restrict__ faces,
    const float* __restrict__ coef, float* __restrict__ out,
    int res, int P, int nfPad)
{
    extern __shared__ float ldsCoef[];

    // Stage all triangle coefficients into LDS (6 floats / tri).
    int total = nfPad * 6;
    for (int i = threadIdx.x; i < total; i += blockDim.x)
        ldsCoef[i] = coef[i];
    __syncthreads();

    int lane   = threadIdx.x & 31;
    int waveId = threadIdx.x >> 5;
    int pixBase = blockIdx.x * 128 + waveId * 16;   // 16 pixels per wave
    int m = lane & 15;
    int p = pixBase + m;

    float invRes = 1.0f / (float)res;
    int x = p % res;
    int y = p / res;

    v2f A;
    if (lane < 16) {
        A[0] = ((float)x + 0.5f) * invRes;   // K=0 row: px
        A[1] = ((float)y + 0.5f) * invRes;   // K=1 row: py
    } else {
        A[0] = 1.0f;                         // K=2 row: constant term
        A[1] = 0.0f;                         // K=3 row: padding
    }

    int   myIdx = -1;
    float myW0 = 0.f, myW1 = 0.f;
    unsigned pending = 0xFFFFu;              // one bit per wave-local pixel

    int nBlocks = nfPad >> 4;
    for (int blk = 0; blk < nBlocks; ++blk) {
        const float* tb = ldsCoef + (blk * 16 + m) * 6;
        v2f B0, B1;
        if (lane < 16) {
            B0[0] = tb[0]; B0[1] = tb[1];    // rows K=0,1 of w0 columns
            B1[0] = tb[3]; B1[1] = tb[4];    // rows K=0,1 of w1 columns
        } else {
            B0[0] = tb[2]; B0[1] = 0.0f;     // rows K=2,3 of w0 columns
            B1[0] = tb[5]; B1[1] = 0.0f;     // rows K=2,3 of w1 columns
        }

        v8f z = {};
        v8f W0 = __builtin_amdgcn_wmma_f32_16x16x4_f32(
            false, A, false, B0, (short)0, z, false, false);
        v8f W1 = __builtin_amdgcn_wmma_f32_16x16x4_f32(
            false, A, false, B1, (short)0, z, false, false);

#pragma unroll
        for (int v = 0; v < 8; ++v) {
            float w0 = W0[v];
            float w1 = W1[v];
            bool inside = (w0 >= 0.f) && (w1 >= 0.f) && ((1.f - w0 - w1) >= 0.f);
            unsigned long long bal = __ballot(inside);
            unsigned mlo = (unsigned)(bal & 0xFFFFull);          // pixel v
            unsigned mhi = (unsigned)((bal >> 16) & 0xFFFFull);  // pixel v+8

            if (((pending >> v) & 1u) && mlo) {
                int f = __builtin_ctz(mlo);
                float fw0 = __shfl(w0, f, 32);
                float fw1 = __shfl(w1, f, 32);
                if (lane == v) { myIdx = blk * 16 + f; myW0 = fw0; myW1 = fw1; }
                pending &= ~(1u << v);
            }
            if (((pending >> (v + 8)) & 1u) && mhi) {
                int f = __builtin_ctz(mhi);
                float fw0 = __shfl(w0, 16 + f, 32);
                float fw1 = __shfl(w1, 16 + f, 32);
                if (lane == v + 8) { myIdx = blk * 16 + f; myW0 = fw0; myW1 = fw1; }
                pending &= ~(1u << (v + 8));
            }
        }
        if (!pending) break;                 // uniform early exit
    }

    // Lanes 0-15 own the 16 pixel states: gather attributes and store.
    if (lane < 16 && p < P) {
        float o0 = 0.f, o1 = 0.f, o2 = 0.f;
        if (myIdx >= 0) {
            int i0 = faces[3 * myIdx + 0];
            int i1 = faces[3 * myIdx + 1];
            int i2 = faces[3 * myIdx + 2];
            float w2 = 1.f - myW0 - myW1;
            o0 = myW0 * attr[3 * i0 + 0] + myW1 * attr[3 * i1 + 0] + w2 * attr[3 * i2 + 0];
            o1 = myW0 * attr[3 * i0 + 1] + myW1 * attr[3 * i1 + 1] + w2 * attr[3 * i2 + 1];
            o2 = myW0 * attr[3 * i0 + 2] + myW1 * attr[3 * i1 + 2] + w2 * attr[3 * i2 + 2];
        }
        out[3 * p + 0] = o0;
        out[3 * p + 1] = o1;
        out[3 * p + 2] = o2;
    }
}

extern "C" void kernel_launch(void* const* d_in, const int* in_sizes, int n_in,
                              void* d_out, int out_size, void* d_ws, size_t ws_size,
                              hipStream_t stream) {
    const float* attr  = (const float*)d_in[0];
    const float* uv    = (const float*)d_in[1];
    const int*   faces = (const int*)d_in[2];
    float*       out   = (float*)d_out;

    int nf    = in_sizes[2] / 3;
    int nfPad = (nf + 15) & ~15;
    int res   = (int)(sqrt((double)(out_size / 3)) + 0.5);
    int P     = res * res;

    float* coef = (float*)d_ws;              // 6 * nfPad floats of scratch

    int prepBlocks = (nfPad + 255) / 256;
    bake_prep_coeffs<<<prepBlocks, 256, 0, stream>>>(uv, faces, coef, nf, nfPad);

    int mainBlocks = (P + 127) / 128;        // 8 waves * 16 pixels per block
    size_t shmem = (size_t)nfPad * 6 * sizeof(float);
    bake_wmma_raster<<<mainBlocks, 256, shmem, stream>>>(
        attr, faces, coef, out, res, P, nfPad);
}